// NN_26319559590678
// MI455X (gfx1250) — compile-verified
//
#include <hip/hip_runtime.h>
#include <hip/hip_bf16.h>
#include <math.h>

#define SEQ 20
#define H   128
#define H2  256
#define G1  512     // 4*H
#define G2  1024    // 8*H
#define NC  100000

#define NT    ((NC + 127) / 128)   // 782 class tiles of 128
#define NBLK3 256                  // persistent blocks for kernel 3
#define WLSTR 260                  // padded LDS row stride (floats)
#define WLBUF (128 * WLSTR)        // one Wlin tile buffer (floats)

typedef float v2f __attribute__((ext_vector_type(2)));
typedef float v8f __attribute__((ext_vector_type(8)));
typedef unsigned int u32x4 __attribute__((ext_vector_type(4)));
typedef int i32x4 __attribute__((ext_vector_type(4)));
typedef int i32x8 __attribute__((ext_vector_type(8)));

// workspace layout (float offsets)
#define WS_FWD 0
#define WS_BWD (SEQ*H)             // 2560
#define WS_FB  (2*SEQ*H)           // 5120  (hl: [20][256])
#define WS_BAR (WS_FB + SEQ*H2)    // 10240 (one int, barrier counter)

__device__ __forceinline__ float sigmoidf_(float x) { return 1.0f / (1.0f + __expf(-x)); }

// ---------------------------------------------------------------------------
// Kernel 1: forward (block 0) and backward (block 1) LSTM, H=128.
// x-projection hoisted out of the recurrence; Whh^T resident in LDS.
// ---------------------------------------------------------------------------
__global__ __launch_bounds__(512) void k_lstm_fb(
    const float* __restrict__ x,
    const float* __restrict__ h0f, const float* __restrict__ c0f,
    const float* __restrict__ h0b, const float* __restrict__ c0b,
    const float* __restrict__ Wih_f, const float* __restrict__ Whh_f,
    const float* __restrict__ bih_f, const float* __restrict__ bhh_f,
    const float* __restrict__ Wih_b, const float* __restrict__ Whh_b,
    const float* __restrict__ bih_b, const float* __restrict__ bhh_b,
    float* __restrict__ ws)
{
  extern __shared__ float sm[];
  float* WhhT  = sm;                    // [128][513] padded (conflict-free)
  float* xproj = WhhT + 128*513;        // [20][512]
  float* xT    = xproj + SEQ*G1;        // [128][20]
  float* hL    = xT + H*SEQ;            // [128]
  float* cL    = hL + H;                // [128]
  float* gL    = cL + H;                // [512]

  const int dir = blockIdx.x;
  const int tid = threadIdx.x;
  const float* Wih = dir ? Wih_b : Wih_f;
  const float* Whh = dir ? Whh_b : Whh_f;
  const float* bih = dir ? bih_b : bih_f;
  const float* bhh = dir ? bhh_b : bhh_f;
  const float* h0  = dir ? h0b : h0f;
  const float* c0  = dir ? c0b : c0f;
  float* hout = ws + (dir ? WS_BWD : WS_FWD);

  if (dir == 0 && tid == 0) ((int*)(ws + WS_BAR))[0] = 0;  // reset barrier for kernel 2

  // stage x (reversed for backward direction), transposed for broadcast reads
  for (int i = tid; i < SEQ*H; i += 512) {
    int t = i >> 7, k = i & 127;
    int ts = dir ? (SEQ-1-t) : t;
    xT[k*SEQ + t] = x[ts*H + k];
  }
  // stage Whh^T with pad-513 stride (coalesced global read, conflict-light LDS write)
  for (int i = tid; i < G1*H; i += 512) {
    int j = i >> 7, k = i & 127;
    WhhT[k*513 + j] = Whh[i];
  }
  if (tid < H) { hL[tid] = h0[tid]; cL[tid] = c0[tid]; }
  __syncthreads();

  // phase 1: xproj[t][j] = bih[j]+bhh[j] + sum_k x_t[k]*Wih[j][k]  (register-tiled over t)
  {
    float acc[SEQ];
    #pragma unroll
    for (int t = 0; t < SEQ; ++t) acc[t] = 0.f;
    const float* wrow = Wih + tid*H;      // each thread walks a contiguous row
    for (int k = 0; k < H; ++k) {
      float w = wrow[k];
      #pragma unroll
      for (int t = 0; t < SEQ; ++t) acc[t] = fmaf(xT[k*SEQ + t], w, acc[t]);
    }
    float b = bih[tid] + bhh[tid];
    #pragma unroll
    for (int t = 0; t < SEQ; ++t) xproj[t*G1 + tid] = acc[t] + b;
  }
  __syncthreads();

  // phase 2: 20-step recurrence, Whh^T from LDS, h broadcast from LDS
  for (int t = 0; t < SEQ; ++t) {
    float s = xproj[t*G1 + tid];
    for (int k = 0; k < H; ++k) s = fmaf(hL[k], WhhT[k*513 + tid], s);
    gL[tid] = s;
    __syncthreads();
    if (tid < H) {
      float ig = sigmoidf_(gL[tid]);
      float fg = sigmoidf_(gL[H   + tid]);
      float gg = tanhf   (gL[2*H + tid]);
      float og = sigmoidf_(gL[3*H + tid]);
      float c = fg*cL[tid] + ig*gg;
      float h = og*tanhf(c);
      cL[tid] = c; hL[tid] = h;
      hout[t*H + tid] = h;
    }
    __syncthreads();
  }
}

// ---------------------------------------------------------------------------
// Kernel 2: combined LSTM (2H=256), 2 MB of weights split across 16 blocks.
// Block b owns 32 hidden units (all 4 gate rows for them -> 256 KB in LDS).
// Per step: h written to a per-step slot of ws_fb, agent-scope atomic barrier.
// ---------------------------------------------------------------------------
__global__ __launch_bounds__(256) void k_lstm_comb(
    const float* __restrict__ h0l, const float* __restrict__ c0l,
    const float* __restrict__ Wih_l, const float* __restrict__ Whh_l,
    const float* __restrict__ bih_l, const float* __restrict__ bhh_l,
    float* __restrict__ ws)
{
  extern __shared__ float sm[];
  float* WTih  = sm;                    // [256][129] padded
  float* WThh  = WTih + 256*129;        // [256][129]
  float* combL = WThh + 256*129;        // [20][256]
  float* hprev = combL + SEQ*H2;        // [256]
  float* psum  = hprev + H2;            // [256]
  float* gate  = psum + H2;             // [128]
  float* bias  = gate + 128;            // [128]
  float* cL    = bias + 128;            // [32]

  const int b   = blockIdx.x;           // owns units [32b, 32b+32)
  const int tid = threadIdx.x;
  int*   bar = (int*)(ws + WS_BAR);
  float* fb  = ws + WS_FB;

  // local gate-row r = g*32 + u  ->  global row R = 256*g + 32*b + u
  for (int i = tid; i < 128*H2; i += 256) {
    int r = i >> 8, k = i & 255;
    int g = r >> 5, u = r & 31;
    int R = 256*g + 32*b + u;
    WTih[k*129 + r] = Wih_l[R*H2 + k];
    WThh[k*129 + r] = Whh_l[R*H2 + k];
  }
  if (tid < 128) {
    int g = tid >> 5, u = tid & 31;
    int R = 256*g + 32*b + u;
    bias[tid] = bih_l[R] + bhh_l[R];
  }
  for (int i = tid; i < SEQ*H2; i += 256) {
    int t = i >> 8, k = i & 255;
    combL[i] = (k < H) ? ws[WS_FWD + t*H + k] : ws[WS_BWD + t*H + (k - H)];
  }
  if (tid < 32) cL[tid] = c0l[32*b + tid];
  __syncthreads();

  for (int t = 0; t < SEQ; ++t) {
    if (tid < H2) hprev[tid] = (t == 0) ? h0l[tid] : fb[(t-1)*H2 + tid];
    __syncthreads();
    {
      // waves 0-3: input half (combL), waves 4-7: recurrent half (hprev)
      const float* WT = (tid < 128) ? WTih : WThh;
      const float* v  = (tid < 128) ? (combL + t*H2) : hprev;
      int r = tid & 127;
      float s = 0.f;
      for (int k = 0; k < H2; ++k) s = fmaf(WT[k*129 + r], v[k], s);
      psum[tid] = s;
    }
    __syncthreads();
    if (tid < 128) gate[tid] = psum[tid] + psum[128 + tid] + bias[tid];
    __syncthreads();
    if (tid < 32) {
      float ig = sigmoidf_(gate[tid]);
      float fg = sigmoidf_(gate[32 + tid]);
      float gg = tanhf   (gate[64 + tid]);
      float og = sigmoidf_(gate[96 + tid]);
      float c = fg*cL[tid] + ig*gg;
      float h = og*tanhf(c);
      cL[tid] = c;
      fb[t*H2 + 32*b + tid] = h;
    }
    if (t < SEQ-1) {
      __threadfence();
      __syncthreads();
      if (tid == 0)
        __hip_atomic_fetch_add(bar, 1, __ATOMIC_RELEASE, __HIP_MEMORY_SCOPE_AGENT);
      while (__hip_atomic_load(bar, __ATOMIC_ACQUIRE, __HIP_MEMORY_SCOPE_AGENT) < 16*(t+1)) {
        __builtin_amdgcn_s_sleep(1);
      }
      __syncthreads();
    }
  }
}

// ---------------------------------------------------------------------------
// Kernel 3: out[20,100000] = fb[20,256] @ Wlin^T + blin, fp32 WMMA 16x16x4.
// 256 persistent blocks, each streams 128-class tiles of Wlin. Staging is done
// by the Tensor Data Mover (2D descriptor, 256x128 tile, LDS padding 4 DWORDs
// per 256 DWORDs -> stride-260 conflict-free layout, tensor_dim1=rows-left so
// the tail tile is hardware zero-filled), double-buffered with TENSORcnt.
// ---------------------------------------------------------------------------
#define HAVE_TDM __has_builtin(__builtin_amdgcn_tensor_load_to_lds)

#if HAVE_TDM
__device__ __forceinline__ void tdm_load_tile(const float* gbase, int rows_left,
                                              unsigned lds_byte_off) {
  unsigned long long ga = (unsigned long long)(uintptr_t)gbase;
  u32x4 g0;
  g0[0] = 1u;                                       // count=1, user descriptor
  g0[1] = lds_byte_off;                             // D#.lds_addr
  g0[2] = (unsigned)(ga & 0xffffffffu);             // global_addr[31:0]
  g0[3] = (unsigned)((ga >> 32) & 0x1ffffffu)       // global_addr[56:32]
        | (2u << 30);                               // type=2 (image)
  unsigned rows = (unsigned)rows_left;
  i32x8 g1;
  g1[0] = (int)((2u << 16)      // data_size = 4 bytes
        | (1u << 20)            // pad_enable
        | (7u << 22)            // pad_interval: 256 DWORDs
        | (3u << 25));          // pad_amount: 4 DWORDs  -> LDS row stride 260 fl
  g1[1] = (int)(256u << 16);                        // tensor_dim0 = 256 (lo16)
  g1[2] = (int)((rows & 0xffffu) << 16);            // dim0 hi=0 | tensor_dim1 lo16
  g1[3] = (int)((rows >> 16) | (256u << 16));       // tensor_dim1 hi | tile_dim0=256
  g1[4] = 128;                                      // tile_dim1=128, tile_dim2=0
  g1[5] = 256;                                      // tensor_dim0_stride = 256
  g1[6] = 0;
  g1[7] = 0;
  i32x4 z4 = {0, 0, 0, 0};
#if __clang_major__ >= 23
  i32x8 z8 = {0, 0, 0, 0, 0, 0, 0, 0};
  __builtin_amdgcn_tensor_load_to_lds(g0, g1, z4, z4, z8, 0);
#else
  __builtin_amdgcn_tensor_load_to_lds(g0, g1, z4, z4, 0);
#endif
}
#endif

__device__ __forceinline__ void gemm_tile(const float* __restrict__ wl,
                                          const float* __restrict__ fbp,
                                          const float* __restrict__ blin,
                                          float* __restrict__ out,
                                          int n0, int wave, int half, int l16) {
  v8f c0 = {}; v8f c1 = {};
  const float* arow0 = fbp + l16*WLSTR        + 2*half;   // A rows 0-15
  const float* arow1 = fbp + (16 + l16)*WLSTR + 2*half;   // A rows 16-31 (padded)
  const float* brow  = wl  + (wave*16 + l16)*WLSTR + 2*half;
  #pragma unroll 4
  for (int k0 = 0; k0 < H2; k0 += 4) {
    v2f a0 = *(const v2f*)(arow0 + k0);
    v2f a1 = *(const v2f*)(arow1 + k0);
    v2f bb = *(const v2f*)(brow  + k0);
    c0 = __builtin_amdgcn_wmma_f32_16x16x4_f32(false, a0, false, bb, (short)0, c0, false, false);
    c1 = __builtin_amdgcn_wmma_f32_16x16x4_f32(false, a1, false, bb, (short)0, c1, false, false);
  }
  int gn = n0 + wave*16 + l16;
  if (gn < NC) {
    float bv = blin[gn];
    #pragma unroll
    for (int r = 0; r < 8; ++r) {
      int m0 = r + 8*half;                    // C layout: vgpr r, lanes16-31 -> M=r+8
      out[(size_t)m0*NC + gn] = c0[r] + bv;
      int m1 = 16 + r + 8*half;
      if (m1 < SEQ) out[(size_t)m1*NC + gn] = c1[r] + bv;
    }
  }
}

__global__ __launch_bounds__(256) void k_linear_wmma(
    const float* __restrict__ Wlin, const float* __restrict__ blin,
    const float* __restrict__ ws, float* __restrict__ out)
{
  extern __shared__ float sm[];
  float* fbp = sm;                      // [32][260]
  float* wlb = fbp + 32*WLSTR;          // 2 x [128][260] tile buffers
  const float* fb = ws + WS_FB;
  const int tid = threadIdx.x;

  for (int i = tid; i < 32*H2; i += 256) {
    int m = i >> 8, k = i & 255;
    fbp[m*WLSTR + k] = (m < SEQ) ? fb[m*H2 + k] : 0.f;
  }

  const int wave = tid >> 5;
  const int lane = tid & 31;
  const int half = lane >> 4;           // lanes 16-31 carry K+2 / M+8 halves
  const int l16  = lane & 15;

#if HAVE_TDM
  const int tile0 = blockIdx.x;
  if (wave == 0 && tile0 < NT)
    tdm_load_tile(Wlin + (size_t)tile0 * 128 * H2, NC - tile0*128,
                  (unsigned)(32*WLSTR*4));
  int buf = 0;
  for (int tile = tile0; tile < NT; tile += NBLK3) {
    int nxt = tile + NBLK3;
    if (wave == 0) {
      if (nxt < NT) {
        tdm_load_tile(Wlin + (size_t)nxt * 128 * H2, NC - nxt*128,
                      (unsigned)((32*WLSTR + (buf^1)*WLBUF) * 4));
        __builtin_amdgcn_s_wait_tensorcnt((short)1);  // current tile done, prefetch in flight
      } else {
        __builtin_amdgcn_s_wait_tensorcnt((short)0);
      }
    }
    __syncthreads();                                  // tile data visible to all waves
    gemm_tile(wlb + buf*WLBUF, fbp, blin, out, tile*128, wave, half, l16);
    __syncthreads();                                  // done reading before buffer reuse
    buf ^= 1;
  }
#else
  for (int tile = blockIdx.x; tile < NT; tile += NBLK3) {
    int n0 = tile * 128;
    for (int i = tid*4; i < 128*H2; i += 256*4) {
      int n = i >> 8, k = i & 255;
      int gn = n0 + n;
      float4 v = make_float4(0.f, 0.f, 0.f, 0.f);
      if (gn < NC) v = *(const float4*)(Wlin + (size_t)gn*H2 + k);
      *(float4*)(wlb + n*WLSTR + k) = v;
    }
    __syncthreads();
    gemm_tile(wlb, fbp, blin, out, n0, wave, half, l16);
    __syncthreads();
  }
#endif
}

// ---------------------------------------------------------------------------
extern "C" void kernel_launch(void* const* d_in, const int* in_sizes, int n_in,
                              void* d_out, int out_size, void* d_ws, size_t ws_size,
                              hipStream_t stream) {
  (void)in_sizes; (void)n_in; (void)out_size; (void)ws_size;
  const float* x     = (const float*)d_in[0];
  const float* h0f   = (const float*)d_in[1];
  const float* c0f   = (const float*)d_in[2];
  const float* h0b   = (const float*)d_in[3];
  const float* c0b   = (const float*)d_in[4];
  const float* h0l   = (const float*)d_in[5];
  const float* c0l   = (const float*)d_in[6];
  const float* Wih_f = (const float*)d_in[7];
  const float* Whh_f = (const float*)d_in[8];
  const float* bih_f = (const float*)d_in[9];
  const float* bhh_f = (const float*)d_in[10];
  const float* Wih_b = (const float*)d_in[11];
  const float* Whh_b = (const float*)d_in[12];
  const float* bih_b = (const float*)d_in[13];
  const float* bhh_b = (const float*)d_in[14];
  const float* Wih_l = (const float*)d_in[15];
  const float* Whh_l = (const float*)d_in[16];
  const float* bih_l = (const float*)d_in[17];
  const float* bhh_l = (const float*)d_in[18];
  const float* Wlin  = (const float*)d_in[19];
  const float* blin  = (const float*)d_in[20];
  float* out = (float*)d_out;
  float* ws  = (float*)d_ws;

  const size_t sm1 = (size_t)(128*513 + SEQ*G1 + H*SEQ + H + H + G1) * sizeof(float);         // ~317 KB
  const size_t sm2 = (size_t)(2*256*129 + SEQ*H2 + H2 + H2 + 128 + 128 + 32) * sizeof(float); // ~288 KB
  const size_t sm3 = (size_t)(32*WLSTR + 2*WLBUF) * sizeof(float);                            // ~293 KB

  k_lstm_fb<<<dim3(2), dim3(512), sm1, stream>>>(x, h0f, c0f, h0b, c0b,
      Wih_f, Whh_f, bih_f, bhh_f, Wih_b, Whh_b, bih_b, bhh_b, ws);
  k_lstm_comb<<<dim3(16), dim3(256), sm2, stream>>>(h0l, c0l, Wih_l, Whh_l, bih_l, bhh_l, ws);
  k_linear_wmma<<<dim3(NBLK3), dim3(256), sm3, stream>>>(Wlin, blin, ws, out);
}